// GCNPlusRConv_27419071218310
// MI455X (gfx1250) — compile-verified
//
#include <hip/hip_runtime.h>
#include <hip/hip_bf16.h>

typedef float v2f __attribute__((ext_vector_type(2)));
typedef float v8f __attribute__((ext_vector_type(8)));

#define D_IN   128
#define D_OUT  32
#define K_HOPS 10

// ---------------------------------------------------------------------------
// h = x @ W + bias   via V_WMMA_F32_16X16X4_F32 (fp32 in, fp32 accum)
// One wave -> one 16x16 tile of h. Block = 64 threads = 2 waves = 16x32 rows.
// A 16x4 f32 layout: lanes 0-15: M=lane, K={k,k+1}; lanes 16-31: M=lane-16, K={k+2,k+3}
// B 4x16 f32 layout: lanes 0-15: N=lane, K={k,k+1}; lanes 16-31: N=lane-16, K={k+2,k+3}
// C/D: VGPR r -> M=r (lanes 0-15, N=lane) / M=r+8 (lanes 16-31, N=lane-16)
// ---------------------------------------------------------------------------
__global__ void tdconv_gemm_bias(const float* __restrict__ x,
                                 const float* __restrict__ W,
                                 const float* __restrict__ bias,
                                 float* __restrict__ h, int N) {
  const int wave = threadIdx.x >> 5;        // 0..1 -> column tile
  const int lane = threadIdx.x & 31;
  const int r0   = blockIdx.x << 4;
  const int c0   = wave << 4;
  const int half = lane >> 4;               // 0 or 1
  const int l    = lane & 15;
  const int kofs = half << 1;               // 0 or 2

  int arow = r0 + l;
  if (arow >= N) arow = N - 1;              // clamp loads, keep EXEC all-ones
  const float* xrow = x + (size_t)arow * D_IN + kofs;
  const int bcol = c0 + l;

  v8f acc = {0.f, 0.f, 0.f, 0.f, 0.f, 0.f, 0.f, 0.f};
#pragma unroll
  for (int k = 0; k < D_IN; k += 4) {
    v2f a, b;
    a.x = xrow[k];
    a.y = xrow[k + 1];
    b.x = W[(k + kofs) * D_OUT + bcol];
    b.y = W[(k + kofs + 1) * D_OUT + bcol];
    acc = __builtin_amdgcn_wmma_f32_16x16x4_f32(false, a, false, b,
                                                (short)0, acc, false, false);
  }
  const float bv = bias[bcol];
#pragma unroll
  for (int r = 0; r < 8; ++r) {
    const int row = r0 + r + (half << 3);
    if (row < N) h[(size_t)row * D_OUT + bcol] = acc[r] + bv;
  }
}

// ---------------------------------------------------------------------------
// Normalization precompute
// ---------------------------------------------------------------------------
__global__ void k_deg_init(float* __restrict__ deg, int N) {
  int i = blockIdx.x * blockDim.x + threadIdx.x;
  if (i < N) deg[i] = 1.0f;                 // self-loop
}

__global__ void k_deg_count(const int* __restrict__ dst, float* __restrict__ deg, int E) {
  int e = blockIdx.x * blockDim.x + threadIdx.x;
  if (e < E) unsafeAtomicAdd(&deg[dst[e]], 1.0f);
}

__global__ void k_rsqrt_inplace(float* __restrict__ deg, int N) {
  int i = blockIdx.x * blockDim.x + threadIdx.x;
  if (i < N) deg[i] = rsqrtf(deg[i]);       // deg >= 1 always (self-loops)
}

__global__ void k_wnorm(const int* __restrict__ src, const int* __restrict__ dst,
                        const float* __restrict__ dinv, float* __restrict__ wn, int E) {
  int e = blockIdx.x * blockDim.x + threadIdx.x;
  if (e < E) wn[e] = dinv[src[e]] * dinv[dst[e]];
}

// Taylor coefficients e^{-t} t^k / k!  computed on-device (t is a device scalar)
__global__ void k_coeffs(const float* __restrict__ t, float* __restrict__ cf) {
  if (blockIdx.x == 0 && threadIdx.x == 0) {
    float tv = t[0];
    float c = expf(-tv);
    cf[0] = c;
    for (int k = 1; k <= K_HOPS; ++k) {
      c = c * tv / (float)k;
      cf[k] = c;
    }
  }
}

// ---------------------------------------------------------------------------
// Diffusion loop kernels
// ---------------------------------------------------------------------------
__global__ void k_scale_out(const float* __restrict__ h, const float* __restrict__ cf,
                            float* __restrict__ out, int n) {
  int i = blockIdx.x * blockDim.x + threadIdx.x;
  if (i < n) out[i] = cf[0] * h[i];
}

// next = (dinv[i]^2) * cur   (self-loop term, plain write -> initializes next)
__global__ void k_self(const float* __restrict__ cur, const float* __restrict__ dinv,
                       float* __restrict__ next, int n) {
  int i = blockIdx.x * blockDim.x + threadIdx.x;
  if (i < n) {
    float dv = dinv[i >> 5];                // D_OUT == 32
    next[i] = dv * dv * cur[i];
  }
}

// scatter-add over edges: 4 threads per edge, 8 channels each (float4 gathers)
__global__ void k_edge(const int* __restrict__ src, const int* __restrict__ dst,
                       const float* __restrict__ wn, const float* __restrict__ cur,
                       float* __restrict__ next, int E) {
  int t = blockIdx.x * blockDim.x + threadIdx.x;
  if (t >= (E << 2)) return;
  int e = t >> 2;
  int g = t & 3;
  int s = src[e];
  int d = dst[e];
  float w = wn[e];
  const float4* cs = (const float4*)(cur + ((size_t)s << 5) + (g << 3));
  float4 v0 = cs[0];
  float4 v1 = cs[1];
  float* nd = next + ((size_t)d << 5) + (g << 3);
  unsafeAtomicAdd(nd + 0, w * v0.x);
  unsafeAtomicAdd(nd + 1, w * v0.y);
  unsafeAtomicAdd(nd + 2, w * v0.z);
  unsafeAtomicAdd(nd + 3, w * v0.w);
  unsafeAtomicAdd(nd + 4, w * v1.x);
  unsafeAtomicAdd(nd + 5, w * v1.y);
  unsafeAtomicAdd(nd + 6, w * v1.z);
  unsafeAtomicAdd(nd + 7, w * v1.w);
}

__global__ void k_axpy(const float* __restrict__ v, const float* __restrict__ cf,
                       int k, float* __restrict__ out, int n) {
  int i = blockIdx.x * blockDim.x + threadIdx.x;
  if (i < n) out[i] += cf[k] * v[i];
}

// ---------------------------------------------------------------------------
extern "C" void kernel_launch(void* const* d_in, const int* in_sizes, int n_in,
                              void* d_out, int out_size, void* d_ws, size_t ws_size,
                              hipStream_t stream) {
  const float* x    = (const float*)d_in[0];
  const int*   ei   = (const int*)d_in[1];   // [2, E] flat: src then dst (int32)
  const float* W    = (const float*)d_in[2];
  const float* bias = (const float*)d_in[3];
  const float* t    = (const float*)d_in[4];
  float*       out  = (float*)d_out;

  const int N = in_sizes[0] / D_IN;
  const int E = in_sizes[1] / 2;
  const int* src = ei;
  const int* dst = ei + E;

  // workspace carve-out (256B aligned)
  char* ws = (char*)d_ws;
  size_t off = 0;
  auto carve = [&](size_t nbytes) -> char* {
    char* p = ws + off;
    off += (nbytes + 255) & ~(size_t)255;
    return p;
  };
  float* bufA = (float*)carve((size_t)N * D_OUT * sizeof(float));  // h / ping
  float* bufB = (float*)carve((size_t)N * D_OUT * sizeof(float));  // pong
  float* deg  = (float*)carve((size_t)N * sizeof(float));          // deg -> dinv
  float* wn   = (float*)carve((size_t)E * sizeof(float));          // per-edge norm
  float* cf   = (float*)carve(64);                                 // 11 coeffs
  (void)ws_size;

  const int ne = N * D_OUT;
  const int TB = 256;

  // h = x@W + b  (WMMA)
  tdconv_gemm_bias<<<(N + 15) / 16, 64, 0, stream>>>(x, W, bias, bufA, N);
  // coefficients from device scalar t
  k_coeffs<<<1, 32, 0, stream>>>(t, cf);
  // GCN normalization
  k_deg_init<<<(N + TB - 1) / TB, TB, 0, stream>>>(deg, N);
  k_deg_count<<<(E + TB - 1) / TB, TB, 0, stream>>>(dst, deg, E);
  k_rsqrt_inplace<<<(N + TB - 1) / TB, TB, 0, stream>>>(deg, N);
  k_wnorm<<<(E + TB - 1) / TB, TB, 0, stream>>>(src, dst, deg, wn, E);
  // out = cf[0] * h
  k_scale_out<<<(ne + TB - 1) / TB, TB, 0, stream>>>(bufA, cf, out, ne);

  float* cur = bufA;
  float* nxt = bufB;
  for (int k = 1; k <= K_HOPS; ++k) {
    k_self<<<(ne + TB - 1) / TB, TB, 0, stream>>>(cur, deg, nxt, ne);
    k_edge<<<((E << 2) + TB - 1) / TB, TB, 0, stream>>>(src, dst, wn, cur, nxt, E);
    k_axpy<<<(ne + TB - 1) / TB, TB, 0, stream>>>(nxt, cf, k, out, ne);
    float* tmp = cur; cur = nxt; nxt = tmp;
  }
}